// ContrastiveLoss_85306640433695
// MI455X (gfx1250) — compile-verified
//
#include <hip/hip_runtime.h>
#include <math.h>

typedef __attribute__((ext_vector_type(2))) float v2f;
typedef __attribute__((ext_vector_type(8))) float v8f;

#define NHALF 4096
#define DIM   256
#define INV_TEMP 0.5f
#define EPSV 1e-8f
#define NB32  256   // 8192 / 32 blocks of 32 rows per side

// ---------------------------------------------------------------------------
// ws layout:
//   [0, 8192) floats                : row norms
//   byte offset 32768: double[2]    : { sim_all, sim_s } accumulators
// ---------------------------------------------------------------------------

__global__ void cl_init_accum(double* __restrict__ acc) {
    if (threadIdx.x < 2) acc[threadIdx.x] = 0.0;
}

// One wave (32 lanes) per row: coalesced strided loads + shuffle reduction.
__global__ void cl_row_norms(const float* __restrict__ x, float* __restrict__ nrm) {
    const int wave = threadIdx.x >> 5;
    const int lane = threadIdx.x & 31;
    const int row  = blockIdx.x * 8 + wave;           // grid = 1024 -> 8192 rows
    const float* rp = x + (size_t)row * DIM;
    float s = 0.f;
    #pragma unroll
    for (int k = lane; k < DIM; k += 32) {
        float v = rp[k];
        s += v * v;
    }
    #pragma unroll
    for (int off = 16; off; off >>= 1) s += __shfl_xor(s, off, 32);
    if (lane == 0) nrm[row] = sqrtf(s);
}

// Gram kernel: block = 256 threads = 8 waves; each wave computes a 32x32
// output region as a 2x2 grid of 16x16 WMMA tiles (register blocking:
// 4 fragment loads feed 4 v_wmma_f32_16x16x4_f32 per K-step).
__global__ void cl_gram_loss(const float* __restrict__ x,
                             const float* __restrict__ nrm,
                             double* __restrict__ acc) {
    const int wave  = threadIdx.x >> 5;
    const int lane  = threadIdx.x & 31;
    const int tb32  = blockIdx.x;                 // 0..255 (32-col block)
    const int ta32  = blockIdx.y * 8 + wave;      // 0..255 (32-row block)

    double sAll = 0.0, sS = 0.0;

    // Wave-uniform triangular skip at 32x32 granularity (exact: 4096 % 32 == 0,
    // so the fold a&4095 never splits a 32-block).
    const bool active = ((ta32 & (NB32 / 2 - 1)) <= (tb32 & (NB32 / 2 - 1)));
    if (active) {
        const int a_base = ta32 * 32;
        const int b_base = tb32 * 32;
        const int m    = lane & 15;          // row within 16-tile
        const int koff = (lane >> 4) * 2;    // K sub-offset per ISA A/B layout

        const float* a0p = x + (size_t)(a_base + m)      * DIM + koff;
        const float* a1p = x + (size_t)(a_base + 16 + m) * DIM + koff;
        const float* b0p = x + (size_t)(b_base + m)      * DIM + koff;
        const float* b1p = x + (size_t)(b_base + 16 + m) * DIM + koff;

        v8f c00 = {}, c01 = {}, c10 = {}, c11 = {};
        for (int kk = 0; kk < DIM; kk += 4) {
            v2f a0 = *reinterpret_cast<const v2f*>(a0p + kk);
            v2f a1 = *reinterpret_cast<const v2f*>(a1p + kk);
            v2f b0 = *reinterpret_cast<const v2f*>(b0p + kk);
            v2f b1 = *reinterpret_cast<const v2f*>(b1p + kk);
            c00 = __builtin_amdgcn_wmma_f32_16x16x4_f32(false, a0, false, b0, (short)0, c00, false, false);
            c01 = __builtin_amdgcn_wmma_f32_16x16x4_f32(false, a0, false, b1, (short)0, c01, false, false);
            c10 = __builtin_amdgcn_wmma_f32_16x16x4_f32(false, a1, false, b0, (short)0, c10, false, false);
            c11 = __builtin_amdgcn_wmma_f32_16x16x4_f32(false, a1, false, b1, (short)0, c11, false, false);
        }

        // Epilogue: C layout -> lane L, VGPR v holds (M = v + 8*(L>=16), N = L%16)
        const int mbase = (lane >> 4) * 8;
        const int ncol  = lane & 15;
        #pragma unroll
        for (int ib = 0; ib < 2; ++ib) {
            const int bcol = b_base + ib * 16 + ncol;
            const float nb = nrm[bcol];
            const int jj   = bcol & (NHALF - 1);
            #pragma unroll
            for (int ia = 0; ia < 2; ++ia) {
                const v8f* cp = (ia == 0) ? (ib == 0 ? &c00 : &c01)
                                          : (ib == 0 ? &c10 : &c11);
                #pragma unroll
                for (int v = 0; v < 8; ++v) {
                    const int aidx  = a_base + ia * 16 + mbase + v;
                    const float na  = nrm[aidx];
                    const int ii    = aidx & (NHALF - 1);
                    const float den = fmaxf(na * nb, EPSV);
                    const float c   = (*cp)[v] / den;
                    const float e   = expf(c * INV_TEMP);
                    if (ii <= jj)                                   sAll += (double)e;
                    if (aidx < NHALF && bcol >= NHALF && ii == jj)  sS   += (double)e;
                }
            }
        }
    }

    // Block reduction in double, one f64 atomic pair per block.
    __shared__ double redA[256];
    __shared__ double redS[256];
    redA[threadIdx.x] = sAll;
    redS[threadIdx.x] = sS;
    __syncthreads();
    #pragma unroll
    for (int s = 128; s; s >>= 1) {
        if (threadIdx.x < s) {
            redA[threadIdx.x] += redA[threadIdx.x + s];
            redS[threadIdx.x] += redS[threadIdx.x + s];
        }
        __syncthreads();
    }
    if (threadIdx.x == 0) {
        atomicAdd(&acc[0], redA[0]);
        atomicAdd(&acc[1], redS[0]);
    }
}

__global__ void cl_finalize(const double* __restrict__ acc, float* __restrict__ out) {
    if (threadIdx.x == 0) {
        // -log(sim_s / sim_all) = log(sim_all) - log(sim_s)
        out[0] = (float)(log(acc[0]) - log(acc[1]));
    }
}

extern "C" void kernel_launch(void* const* d_in, const int* in_sizes, int n_in,
                              void* d_out, int out_size, void* d_ws, size_t ws_size,
                              hipStream_t stream) {
    (void)in_sizes; (void)n_in; (void)out_size; (void)ws_size;
    const float* x   = (const float*)d_in[0];
    float*  nrm      = (float*)d_ws;
    double* acc      = (double*)((char*)d_ws + 8192 * sizeof(float));
    float*  out      = (float*)d_out;

    cl_init_accum<<<1, 32, 0, stream>>>(acc);
    cl_row_norms<<<1024, 256, 0, stream>>>(x, nrm);
    cl_gram_loss<<<dim3(NB32, NB32 / 8), 256, 0, stream>>>(x, nrm, acc);
    cl_finalize<<<1, 32, 0, stream>>>(acc, out);
}